// DiffusionEngine_51333449121793
// MI455X (gfx1250) — compile-verified
//
#include <hip/hip_runtime.h>
#include <math.h>

// MI455X / gfx1250, wave32. One wave per 19x19 board.
// conv3x3(8-neighbor mean) = 0.125*(T*X*T - X), T = tridiagonal ones band (19x19,
// zero-padded to 32x32). Fold scale: Ts = s*T, s^2 = 0.125*sigmoid(alpha), so
// Ts*X*Ts = 0.125*a*boxsum and the step is X <- (1 - 1.125a)X + Ts*X*Ts + bt*src.
//
// All-f16 pipeline using V_WMMA_F16_16X16X32_F16 (f16 C/D):
//   - matmul-1 D tiles ARE matmul-2 A fragments (register concat, zero moves)
//   - matmul-2 D tiles ARE the next step's resident field fragments (zero moves)
//   - cc*f + bt*src is folded into matmul-2's C operand (free add in the matrix op)
//   - gating/ReLU as packed v16h arithmetic (v_pk_fma_f16 / v_pk_max)

typedef __attribute__((ext_vector_type(16))) _Float16 v16h;
typedef __attribute__((ext_vector_type(8)))  _Float16 v8h;

#define GO_N     19
#define GO_CELLS 361
#define WPB      8      // waves (boards) per 256-thread block

__device__ __forceinline__ v8h wmma16h(v16h a, v16h b, v8h c) {
  // D(16x16 f16) = A(16x32 f16) * B(32x16 f16) + C(16x16 f16)
  return __builtin_amdgcn_wmma_f16_16x16x32_f16(false, a, false, b, (short)0, c,
                                                false, false);
}

__device__ __forceinline__ v16h cat8(v8h lo, v8h hi) {
  return __builtin_shufflevector(lo, hi, 0, 1, 2, 3, 4, 5, 6, 7,
                                         8, 9, 10, 11, 12, 13, 14, 15);
}
__device__ __forceinline__ v8h lo8(v16h x) {
  return __builtin_shufflevector(x, x, 0, 1, 2, 3, 4, 5, 6, 7);
}
__device__ __forceinline__ v8h hi8(v16h x) {
  return __builtin_shufflevector(x, x, 8, 9, 10, 11, 12, 13, 14, 15);
}
__device__ __forceinline__ v16h splat16(_Float16 x) {
  v16h r;
#pragma unroll
  for (int j = 0; j < 16; ++j) r[j] = x;
  return r;
}

// fragment half j of row-tile q <-> board element
//   row = lane%16 + 16q , col = (lane<16?0:8) + (j<8 ? j : 8+j)
__device__ __forceinline__ int col_of(int j, int chalf) {
  return (j < 8) ? (chalf + j) : (16 + chalf + (j - 8));
}

// One diffusion step for one field resident as A-fragments fA[0..1].
// btsrc = bt * src (precomputed, fragment layout). ccv = splat(1 - 1.125a).
__device__ __forceinline__ void diffuse_field(v16h (&fA)[2], const v16h (&btsrc)[2],
                                              v16h ccv, v16h B0, v16h B1) {
  v8h zh = {};
  // M1 = X * Ts (row blur, scaled); f16 D tiles
  v8h E00 = wmma16h(fA[0], B0, zh);
  v8h E01 = wmma16h(fA[0], B1, zh);
  v8h E10 = wmma16h(fA[1], B0, zh);
  v8h E11 = wmma16h(fA[1], B1, zh);
  // f16 C/D layout == A-fragment K-chunk layout: concat is the whole transpose
  v16h A20 = cat8(E00, E10);
  v16h A21 = cat8(E01, E11);
  // decay + source, folded into matmul-2's accumulator (fragment layout == Z^T C layout)
  v16h u0 = ccv * fA[0] + btsrc[0];
  v16h u1 = ccv * fA[1] + btsrc[1];
  // Z = M1^T * Ts + C = (scaled 3x3 box sum)^T + (cc*f + bt*src)^T
  fA[0] = cat8(wmma16h(A20, B0, lo8(u0)), wmma16h(A21, B0, hi8(u0)));
  fA[1] = cat8(wmma16h(A20, B1, lo8(u1)), wmma16h(A21, B1, hi8(u1)));
}

__device__ __forceinline__ void dump_field(const v16h (&fA)[2], float* st,
                                           float* __restrict__ dst,
                                           int lane, int rbase, int chalf) {
  // scatter fragment layout -> LDS row-major 19x19, then coalesced global store
#pragma unroll
  for (int q = 0; q < 2; ++q)
#pragma unroll
    for (int j = 0; j < 16; ++j) {
      int r = rbase + 16 * q, c = col_of(j, chalf);
      if (r < GO_N && c < GO_N) st[r * GO_N + c] = (float)fA[q][j];
    }
  asm volatile("s_wait_dscnt 0x0" ::: "memory");  // intra-wave LDS RAW fence
  for (int i = lane; i < GO_CELLS; i += 32) dst[i] = st[i];
}

__global__ __launch_bounds__(256) void go_diffuse_kernel(
    const float* __restrict__ board, const float* __restrict__ alpha_p,
    const float* __restrict__ beta_p, const float* __restrict__ gamma_p,
    float* __restrict__ out, int nB) {
  __shared__ float stage[WPB][384];
  const int lane = threadIdx.x & 31;
  const int wave = threadIdx.x >> 5;
  const int bid  = blockIdx.x * WPB + wave;
  if (bid >= nB) return;  // wave-uniform exit: EXEC stays all-ones for WMMA

  const float a  = 1.f / (1.f + __expf(-alpha_p[0]));
  const float bv = beta_p[0];
  const float bt = (bv > 20.f) ? bv : log1pf(__expf(bv));  // softplus
  const float g  = 1.f / (1.f + __expf(-gamma_p[0]));
  const v16h ccv = splat16((_Float16)(1.f - 1.125f * a));  // (1-a) - 0.125a
  const _Float16 bth = (_Float16)bt;
  const _Float16 gh  = (_Float16)g;
  const float s  = sqrtf(0.125f * a);  // band value: s^2 = 0.125*a
  const _Float16 h0 = (_Float16)0.f;
  const v16h zv = {};

  // coalesced board load -> LDS
  float* st = stage[wave];
  const float* bg = board + (size_t)bid * GO_CELLS;
  for (int i = lane; i < GO_CELLS; i += 32) st[i] = bg[i];
  asm volatile("s_wait_dscnt 0x0" ::: "memory");

  const int rbase = lane & 15;
  const int chalf = (lane < 16) ? 0 : 8;
  const int klo   = (lane < 16) ? 0 : 16;

  // Constant band-operator B fragments (32x16 f16 B, contiguous k order used here: N = lane%16 (+16/tile),
  // lanes>=16 take K+16, vgpr/half j -> K = klo + j).
  v16h B0, B1;
#pragma unroll
  for (int j = 0; j < 16; ++j) {
    int k  = klo + j;
    int d0 = k - rbase;        d0 = d0 < 0 ? -d0 : d0;
    int d1 = k - (rbase + 16); d1 = d1 < 0 ? -d1 : d1;
    B0[j] = (d0 <= 1 && k < GO_N)                      ? (_Float16)s : h0;
    B1[j] = (d1 <= 1 && k < GO_N && rbase + 16 < GO_N) ? (_Float16)s : h0;
  }

  // Gather board; build resident fragments + loop-invariant gated/source terms.
  // src masks == srcs (stones are 0/1): bs = relu(board), ws = relu(-board).
  v16h bfA[2], wfA[2];     // fields (start = sources)
  v16h btb[2], btw[2];     // bt * src
  v16h gb[2],  gw[2];      // g  * src  (gating coefficient)
#pragma unroll
  for (int q = 0; q < 2; ++q)
#pragma unroll
    for (int j = 0; j < 16; ++j) {
      int r = rbase + 16 * q, c = col_of(j, chalf);
      _Float16 bd = (r < GO_N && c < GO_N) ? (_Float16)st[r * GO_N + c] : h0;
      _Float16 bs = bd > h0 ? bd : h0;
      _Float16 ws = bd < h0 ? (_Float16)(-bd) : h0;
      bfA[q][j] = bs;  wfA[q][j] = ws;
      btb[q][j] = bth * bs;  btw[q][j] = bth * ws;
      gb[q][j]  = gh * bs;   gw[q][j]  = gh * ws;
    }

  int sidx = 0;
  for (int t = 1; t <= 10; ++t) {
    diffuse_field(bfA, btb, ccv, B0, B1);
    diffuse_field(wfA, btw, ccv, B0, B1);
    // sequential gating + ReLU, packed f16 vector ops
    // (b sees updated w; w sees gated pre-relu b)
#pragma unroll
    for (int q = 0; q < 2; ++q) {
      v16h b2 = bfA[q] - gb[q] * wfA[q];
      v16h w2 = wfA[q] - gw[q] * b2;
      bfA[q] = __builtin_elementwise_max(b2, zv);
      wfA[q] = __builtin_elementwise_max(w2, zv);
    }
    if (t == 2 || t == 5 || t == 10) {
      float* phib = out + ((size_t)bid * 3 + sidx) * GO_CELLS;
      float* phiw = out + (size_t)nB * 3 * GO_CELLS + ((size_t)bid * 3 + sidx) * GO_CELLS;
      dump_field(bfA, st, phib, lane, rbase, chalf);
      dump_field(wfA, st, phiw, lane, rbase, chalf);
      ++sidx;
    }
  }
}

// k_field_new = 0.7*k_field + ko  (fully coalesced elementwise)
__global__ void k_field_update(const float* __restrict__ kf,
                               const unsigned char* __restrict__ ko,
                               float* __restrict__ dst, long n) {
  long i = (long)blockIdx.x * blockDim.x + threadIdx.x;
  if (i < n) dst[i] = 0.7f * kf[i] + (ko[i] ? 1.0f : 0.0f);
}

extern "C" void kernel_launch(void* const* d_in, const int* in_sizes, int n_in,
                              void* d_out, int out_size, void* d_ws, size_t ws_size,
                              hipStream_t stream) {
  (void)n_in; (void)out_size; (void)d_ws; (void)ws_size;
  const float*         board = (const float*)d_in[0];
  const float*         kf    = (const float*)d_in[1];
  const unsigned char* ko    = (const unsigned char*)d_in[2];  // jax bool = 1 byte
  const float*         alpha = (const float*)d_in[3];
  const float*         beta  = (const float*)d_in[4];
  const float*         gamma = (const float*)d_in[5];
  float* out = (float*)d_out;

  const int nB = in_sizes[0] / GO_CELLS;
  go_diffuse_kernel<<<dim3((nB + WPB - 1) / WPB), 256, 0, stream>>>(
      board, alpha, beta, gamma, out, nB);

  const long nk = (long)nB * GO_CELLS;
  k_field_update<<<dim3((unsigned)((nk + 255) / 256)), 256, 0, stream>>>(
      kf, ko, out + (size_t)nB * 6 * GO_CELLS, nk);
}